// MambaBlock_10960756539579
// MI455X (gfx1250) — compile-verified
//
#include <hip/hip_runtime.h>
#include <cstdint>
#include <cstddef>

// ---------------------------------------------------------------------------
// MI455X (gfx1250) bidirectional Mamba block.
// GEMMs -> V_WMMA_F32_16X16X4_F32 (exact fp32 on the matrix pipe), with
// compile-time strides (immediate-offset addressing) and 4 N-tiles per wave
// (16x64 strip) to amortize A loads: 4 WMMAs per 9 loads per k-step.
// Selective scan -> 3-phase chunked associative scan (16K-wave parallelism).
// All intermediates (~61MB) L2-resident on the 192MB L2.
// ---------------------------------------------------------------------------

typedef float v2f __attribute__((ext_vector_type(2)));
typedef float v8f __attribute__((ext_vector_type(8)));

#define LQ   4096   // sequence length H*W*T
#define BQ   4      // batch
#define CQ   64     // embed dim
#define DIQ  128    // d_inner
#define E2Q  256    // 2*d_inner
#define NSQ  16     // d_state
#define EPQ  48     // padded x_dbl rows (40 valid: 8 dt, 16 B, 16 C)
#define NCHQ 64     // number of scan chunks
#define CHQ  64     // chunk length

// ---------------------------------------------------------------------------
// K0: (B,C,T,H,W) -> seq (B,C,L) with l = (h*16+w)*16 + t
// ---------------------------------------------------------------------------
__global__ void k_transpose_seq(const float* __restrict__ x,
                                float* __restrict__ seq) {
  int idx = blockIdx.x * blockDim.x + threadIdx.x;   // over B*C*L
  int l  = idx & (LQ - 1);
  int bc = idx >> 12;
  int t  = l & 15;
  int hw = l >> 4;
  seq[idx] = x[(size_t)bc * LQ + t * 256 + hw];
}

// ---------------------------------------------------------------------------
// fp32 WMMA GEMM:  D[b][m][n] = sum_k A[m][k] * X[b][k][n]  (+epilogue)
// Compile-time K / strides / postop -> immediate-offset global loads.
// Each wave: one 16x64 output strip = 4 accumulators sharing the A fragment.
// A frag layout (16x4 fp32, ISA 7.12.2): lanes 0-15 K={0,1}, 16-31 K={2,3},
// loaded as one v2f. Partial-M handled branch-free via clamped load + select.
// POSTOP: 0 none, 1 +bias[m], 2 softplus(x + bias[m])
// ---------------------------------------------------------------------------
template <int K, int MVALID,
          long SK, long SL, long XB,      // X strides: k, l, batch
          long DSM, long DSL, long DB,    // D strides: m, l, batch
          int POSTOP, int NT>
__global__ void k_gemm(const float* __restrict__ A,
                       const float* __restrict__ X,
                       float* __restrict__ Dst,
                       const float* __restrict__ bias) {
  constexpr bool PART = (MVALID & 15) != 0;

  int lane = threadIdx.x;
  int half = lane >> 4;
  int lm   = lane & 15;
  int tm = blockIdx.x;
  int tn = blockIdx.y;
  int n0 = tn * (16 * NT);
  int b  = n0 / LQ;
  int l0 = n0 & (LQ - 1);
  const float* Xb = X + (long)b * XB;
  float* Db = Dst + (long)b * DB;

  int row  = tm * 16 + lm;
  bool rOK = !PART || (row < MVALID);
  int rowc = PART ? (rOK ? row : 0) : row;
  const float* Arow = A + (long)rowc * K;

  int nl = l0 + lm;

  v8f acc[NT];
#pragma unroll
  for (int q = 0; q < NT; ++q) acc[q] = (v8f){};

#pragma unroll 4
  for (int k0 = 0; k0 < K; k0 += 4) {
    int ka = k0 + 2 * half;
    v2f av = *(const v2f*)(Arow + ka);          // 8B aligned (ka even, K%4==0)
    if (PART) {                                  // branch-free zero of pad rows
      av.x = rOK ? av.x : 0.f;
      av.y = rOK ? av.y : 0.f;
    }
#pragma unroll
    for (int q = 0; q < NT; ++q) {
      long nn = (long)(nl + q * 16) * SL;
      v2f bv;
      bv.x = Xb[(long)ka * SK + nn];
      bv.y = Xb[(long)(ka + 1) * SK + nn];
      acc[q] = __builtin_amdgcn_wmma_f32_16x16x4_f32(
          false, av, false, bv, (short)0, acc[q], false, false);
    }
  }

#pragma unroll
  for (int q = 0; q < NT; ++q) {
#pragma unroll
    for (int r = 0; r < 8; ++r) {
      int m = tm * 16 + r + 8 * half;            // D layout: VGPR r -> M=r/r+8
      if (PART && m >= MVALID) continue;
      float v = acc[q][r];
      if (POSTOP == 1) {
        v += bias[m];
      } else if (POSTOP == 2) {                  // softplus
        v += bias[m];
        v = (v > 20.f) ? v : __logf(1.f + __expf(v));
      }
      Db[(long)m * DSM + (long)(nl + q * 16) * DSL] = v;
    }
  }
}

// ---------------------------------------------------------------------------
// causal depthwise conv1d (K=4) + SiLU, with optional sequence flip (dir).
// ---------------------------------------------------------------------------
__global__ void k_dwconv_silu(const float* __restrict__ xz,
                              const float* __restrict__ cw,
                              const float* __restrict__ cb,
                              float* __restrict__ xc, int dir) {
  int idx = blockIdx.x * blockDim.x + threadIdx.x;   // over B*DI*L
  int l  = idx & (LQ - 1);
  int bd = idx >> 12;
  int d  = bd & (DIQ - 1);
  int b  = bd >> 7;
  const float* src = xz + ((size_t)b * E2Q + d) * LQ;
  float s = cb[d];
#pragma unroll
  for (int j = 0; j < 4; ++j) {
    int lp = l - 3 + j;
    if (lp >= 0) {
      int ls = dir ? (LQ - 1 - lp) : lp;
      s += cw[d * 4 + j] * src[ls];
    }
  }
  xc[((size_t)b * DIQ + d) * LQ + l] = s / (1.f + __expf(-s));
}

// ---------------------------------------------------------------------------
// Scan phase A: per-(b,d,chunk) aggregates of h <- dA*h + dBu.
// Wave32 covers TWO units (lane halves); lane%16 = state n.
// ---------------------------------------------------------------------------
__global__ void k_scanA(const float* __restrict__ delta,
                        const float* __restrict__ xc,
                        const float* __restrict__ xdbl,
                        const float* __restrict__ A_log,
                        float* __restrict__ aggA, float* __restrict__ aggB) {
  int lane = threadIdx.x;
  int half = lane >> 4, n = lane & 15;
  int unit  = blockIdx.x * 2 + half;       // (b*128+d)*64 + chunk
  int chunk = unit & (NCHQ - 1);
  int bd    = unit >> 6;
  int d = bd & (DIQ - 1), b = bd >> 7;

  float An = -__expf(A_log[d * NSQ + n]);
  const float* dptr = delta + ((size_t)(b * DIQ + d)) * LQ + chunk * CHQ;
  const float* uptr = xc    + ((size_t)(b * DIQ + d)) * LQ + chunk * CHQ;
  const float* row  = xdbl  + ((size_t)b * LQ + chunk * CHQ) * EPQ;

  float h = 0.f, pA = 1.f;
  for (int t = 0; t < CHQ; ++t) {
    __builtin_prefetch(dptr + t + 16, 0, 1);      // global_prefetch path
    float dl = dptr[t];
    float u  = uptr[t];
    float Bm = row[(size_t)t * EPQ + 8 + n];
    float dA = __expf(dl * An);
    h  = dA * h + dl * Bm * u;
    pA *= dA;
  }
  size_t oi = (size_t)unit * NSQ + n;
  aggA[oi] = pA;
  aggB[oi] = h;
}

// ---------------------------------------------------------------------------
// Scan phase B: exclusive prefix over 64 chunk aggregates per (b,d).
// ---------------------------------------------------------------------------
__global__ void k_scanB(const float* __restrict__ aggA,
                        const float* __restrict__ aggB,
                        float* __restrict__ prefix) {
  int lane = threadIdx.x;
  int half = lane >> 4, n = lane & 15;
  int bd = blockIdx.x * 2 + half;          // 0..511
  float pref = 0.f;
  for (int c = 0; c < NCHQ; ++c) {
    size_t oi = (((size_t)bd * NCHQ + c) * NSQ + n);
    prefix[oi] = pref;
    pref = aggA[oi] * pref + aggB[oi];
  }
}

// ---------------------------------------------------------------------------
// Scan phase C: replay chunks seeded with prefixes; reduce y over the 16
// state lanes (width-16 shfl_xor); +u*D; gate with silu(z); accumulate into
// acc at the original (un-flipped) position.
// ---------------------------------------------------------------------------
__global__ void k_scanC(const float* __restrict__ delta,
                        const float* __restrict__ xc,
                        const float* __restrict__ xdbl,
                        const float* __restrict__ xz,
                        const float* __restrict__ A_log,
                        const float* __restrict__ Dp,
                        const float* __restrict__ prefix,
                        float* __restrict__ acc, int dir) {
  int lane = threadIdx.x;
  int half = lane >> 4, n = lane & 15;
  int unit  = blockIdx.x * 2 + half;
  int chunk = unit & (NCHQ - 1);
  int bd    = unit >> 6;
  int d = bd & (DIQ - 1), b = bd >> 7;

  float An = -__expf(A_log[d * NSQ + n]);
  float Dd = Dp[d];
  const float* dptr = delta + ((size_t)(b * DIQ + d)) * LQ + chunk * CHQ;
  const float* uptr = xc    + ((size_t)(b * DIQ + d)) * LQ + chunk * CHQ;
  const float* row  = xdbl  + ((size_t)b * LQ + chunk * CHQ) * EPQ;
  const float* zrow = xz + ((size_t)b * E2Q + DIQ + d) * LQ;

  float h = prefix[(size_t)unit * NSQ + n];
  for (int t = 0; t < CHQ; ++t) {
    float dl = dptr[t];
    float u  = uptr[t];
    float Bm = row[(size_t)t * EPQ + 8 + n];
    float Cm = row[(size_t)t * EPQ + 24 + n];
    float dA = __expf(dl * An);
    h = dA * h + dl * Bm * u;
    float part = h * Cm;
    part += __shfl_xor(part, 8, 16);
    part += __shfl_xor(part, 4, 16);
    part += __shfl_xor(part, 2, 16);
    part += __shfl_xor(part, 1, 16);
    if (n == 0) {
      int l    = chunk * CHQ + t;
      int lorg = dir ? (LQ - 1 - l) : l;           // flip back for output/z
      float y  = part + u * Dd;
      float z  = zrow[lorg];
      float o  = y * (z / (1.f + __expf(-z)));
      size_t oi = ((size_t)(b * DIQ + d)) * LQ + lorg;
      if (dir) acc[oi] += o; else acc[oi] = o;
    }
  }
}

// ---------------------------------------------------------------------------
// 3x3x3 conv (pad 1) + bias + residual add. Output-channel weights staged in
// LDS (6.9KB of the 320KB WGP pool); input streamed from L2.
// ---------------------------------------------------------------------------
__global__ void k_conv3d_res(const float* __restrict__ in,     // [b][ci][4096]
                             const float* __restrict__ w,      // [c][ci][27]
                             const float* __restrict__ bias,
                             const float* __restrict__ resid,  // [b][c][4096]
                             float* __restrict__ out) {
  __shared__ float wsm[64 * 27];
  int c  = blockIdx.x & 63;
  int bt = blockIdx.x >> 6;          // b*16 + t
  int t  = bt & 15, b = bt >> 4;
  for (int i = threadIdx.x; i < 64 * 27; i += 256) wsm[i] = w[(size_t)c * 64 * 27 + i];
  __syncthreads();

  int hw = threadIdx.x;              // 0..255 = one (h,w) plane at fixed t
  int h = hw >> 4, wq = hw & 15;
  float s = bias[c];
  const float* inb = in + (size_t)b * (64 * LQ);
  for (int ci = 0; ci < 64; ++ci) {
    const float* ip = inb + (size_t)ci * LQ;
    const float* wp = wsm + ci * 27;
#pragma unroll
    for (int kt = 0; kt < 3; ++kt) {
      int tt = t + kt - 1;
      if ((unsigned)tt > 15u) continue;
#pragma unroll
      for (int kh = 0; kh < 3; ++kh) {
        int hh = h + kh - 1;
        if ((unsigned)hh > 15u) continue;
#pragma unroll
        for (int kw = 0; kw < 3; ++kw) {
          int ww = wq + kw - 1;
          if ((unsigned)ww > 15u) continue;
          s += wp[(kt * 3 + kh) * 3 + kw] * ip[tt * 256 + hh * 16 + ww];
        }
      }
    }
  }
  size_t oi = ((size_t)(b * 64 + c)) * LQ + t * 256 + hw;
  out[oi] = s + resid[oi];
}

// ---------------------------------------------------------------------------
// Host launch
// ---------------------------------------------------------------------------
extern "C" void kernel_launch(void* const* d_in, const int* in_sizes, int n_in,
                              void* d_out, int out_size, void* d_ws, size_t ws_size,
                              hipStream_t stream) {
  const float* x         = (const float*)d_in[0];
  const float* in_proj_w = (const float*)d_in[1];
  const float* conv_w[2]   = { (const float*)d_in[2],  (const float*)d_in[9]  };
  const float* conv_b[2]   = { (const float*)d_in[3],  (const float*)d_in[10] };
  const float* xproj_w[2]  = { (const float*)d_in[4],  (const float*)d_in[11] };
  const float* dtproj_w[2] = { (const float*)d_in[5],  (const float*)d_in[12] };
  const float* dtproj_b[2] = { (const float*)d_in[6],  (const float*)d_in[13] };
  const float* A_log[2]    = { (const float*)d_in[7],  (const float*)d_in[14] };
  const float* Dvec[2]     = { (const float*)d_in[8],  (const float*)d_in[15] };
  const float* out_proj_w = (const float*)d_in[16];
  const float* proj_w     = (const float*)d_in[17];
  const float* proj_b     = (const float*)d_in[18];
  const float* norm_w     = (const float*)d_in[19];
  const float* norm_b     = (const float*)d_in[20];
  float* out = (float*)d_out;

  float* w = (float*)d_ws;
  // scratch layout (fp32 elements); total ~61 MB, L2-resident
  size_t off = 0;
  float* seq   = w + off; off += (size_t)BQ * CQ  * LQ;          // 4 MB
  float* xz    = w + off; off += (size_t)BQ * E2Q * LQ;          // 16 MB
  float* xc    = w + off; off += (size_t)BQ * DIQ * LQ;          // 8 MB
  float* xdbl  = w + off; off += (size_t)BQ * LQ  * EPQ;         // 3 MB
  float* delta = w + off; off += (size_t)BQ * DIQ * LQ;          // 8 MB
  float* aggA  = w + off; off += (size_t)BQ * DIQ * NCHQ * NSQ;  // 2 MB
  float* aggB  = w + off; off += (size_t)BQ * DIQ * NCHQ * NSQ;  // 2 MB
  float* pref  = w + off; off += (size_t)BQ * DIQ * NCHQ * NSQ;  // 2 MB
  float* acc   = w + off; off += (size_t)BQ * DIQ * LQ;          // 8 MB
  float* recon = w + off; off += (size_t)BQ * LQ  * CQ;          // 4 MB
  float* tmp   = w + off; off += (size_t)BQ * CQ  * LQ;          // 4 MB
  (void)ws_size;

  const dim3 wv(32);
  constexpr int NT = 4;                         // 16x64 strip per wave
  const unsigned gy = (BQ * LQ) / (16 * NT);    // 256 N-strips

  // K0: transpose to sequence layout
  k_transpose_seq<<<dim3((BQ * CQ * LQ) / 256), dim3(256), 0, stream>>>(x, seq);

  // in_proj: xz[b][e][l] = in_proj_w[e][c] * seq[b][c][l]
  k_gemm<CQ, E2Q, LQ, 1, (long)CQ * LQ, LQ, 1, (long)E2Q * LQ, 0, NT>
      <<<dim3(E2Q / 16, gy), wv, 0, stream>>>(in_proj_w, seq, xz, nullptr);

  for (int dir = 0; dir < 2; ++dir) {
    // conv1d + silu (with flip for backward branch)
    k_dwconv_silu<<<dim3((BQ * DIQ * LQ) / 256), dim3(256), 0, stream>>>(
        xz, conv_w[dir], conv_b[dir], xc, dir);

    // xproj: x_dbl[b][l][e(40/48)] = xproj_w[e][d] * xc[b][d][l] (transposed store)
    k_gemm<DIQ, 40, LQ, 1, (long)DIQ * LQ, 1, EPQ, (long)LQ * EPQ, 0, NT>
        <<<dim3(EPQ / 16, gy), wv, 0, stream>>>(xproj_w[dir], xc, xdbl, nullptr);

    // dtproj + softplus: delta[b][d][l] = softplus(dtproj_w[d][r]*dt[b][r][l]+bias)
    k_gemm<8, DIQ, 1, EPQ, (long)LQ * EPQ, LQ, 1, (long)DIQ * LQ, 2, NT>
        <<<dim3(DIQ / 16, gy), wv, 0, stream>>>(dtproj_w[dir], xdbl, delta,
                                                dtproj_b[dir]);

    // chunked associative scan
    k_scanA<<<dim3((BQ * DIQ * NCHQ) / 2), wv, 0, stream>>>(
        delta, xc, xdbl, A_log[dir], aggA, aggB);
    k_scanB<<<dim3((BQ * DIQ) / 2), wv, 0, stream>>>(aggA, aggB, pref);
    k_scanC<<<dim3((BQ * DIQ * NCHQ) / 2), wv, 0, stream>>>(
        delta, xc, xdbl, xz, A_log[dir], Dvec[dir], pref, acc, dir);
  }

  // out_proj: recon[b][l][c] = out_proj_w[c][d] * acc[b][d][l]
  // (transposed store == reference's (B,L,C)->(B,C,T,H,W) reinterpretation)
  k_gemm<DIQ, CQ, LQ, 1, (long)DIQ * LQ, 1, CQ, (long)LQ * CQ, 0, NT>
      <<<dim3(CQ / 16, gy), wv, 0, stream>>>(out_proj_w, acc, recon, nullptr);

  // 3x3x3 conv + bias + residual
  k_conv3d_res<<<dim3(BQ * 16 * 64), dim3(256), 0, stream>>>(
      recon, proj_w, proj_b, x, tmp);

  // final 1x1x1 conv (64x64 GEMM) + bias -> d_out
  k_gemm<CQ, CQ, LQ, 1, (long)CQ * LQ, LQ, 1, (long)CQ * LQ, 1, NT>
      <<<dim3(CQ / 16, gy), wv, 0, stream>>>(norm_w, tmp, out, norm_b);

  (void)in_sizes; (void)n_in; (void)out_size;
}